// SFMStage_10977936408901
// MI455X (gfx1250) — compile-verified
//
#include <hip/hip_runtime.h>
#include <hip/hip_bf16.h>

// ---------------- problem constants ----------------
#define NN 192          // instances
#define CC 256          // instance channels
#define HH 28
#define WW 28
#define HW 784          // 28*28
#define PP 392          // NUM_POINTS
#define NP (NN*PP)      // 75264 points total
#define BB 2
#define CS 256
#define HS 200
#define WS 304
#define HWS (HS*WS)     // 60800
#define PIX (BB*HWS)    // 121600 semantic pixels

typedef float v2f __attribute__((ext_vector_type(2)));
typedef float v8f __attribute__((ext_vector_type(8)));

static __device__ __forceinline__ v8f wmma4(v2f a, v2f b, v8f c) {
    // D = A(16x4,f32) * B(4x16,f32) + C(16x16,f32)
    return __builtin_amdgcn_wmma_f32_16x16x4_f32(false, a, false, b,
                                                 (short)0, c, false, false);
}

// ---------------------------------------------------------------------------
// K1: sem_t[pix][oc] = relu( W_sem[oc,:] . semantic_feat[b,:,y,x] + b_sem[oc] )
// one wave -> 16(out-ch) x 64(pixel) tile: 4 accumulators share each A fetch.
// waves = (256/16) * (121600/64) = 30400 ; 8 waves/block -> 3800 blocks
// consecutive waves in a block share the pixel quad (B loads hit L0).
// ---------------------------------------------------------------------------
__global__ void sem_gemm_kernel(const float* __restrict__ feat,
                                const float* __restrict__ Wsem,
                                const float* __restrict__ bsem,
                                float* __restrict__ semt) {
    const int lane = threadIdx.x & 31;
    const int wave = threadIdx.x >> 5;
    const int tile = blockIdx.x * 8 + wave;
    const int mt   = tile & 15;          // out-channel tile
    const int quad = tile >> 4;          // 64-pixel quad [0,1900)
    const int m  = lane & 15;
    const int hi = lane >> 4;            // 0/1
    const int c0 = mt * 16;

    const float* arow = Wsem + (size_t)(c0 + m) * CS + 2 * hi;

    // per-sub pixel base pointers (quads never straddle images: HWS%64==0)
    const float* fbase[4];
#pragma unroll
    for (int s = 0; s < 4; ++s) {
        const int myPix = quad * 64 + s * 16 + m;
        const int bImg  = myPix / HWS;
        const int pOff  = myPix - bImg * HWS;
        fbase[s] = feat + (size_t)bImg * CS * HWS + pOff;
    }

    v8f acc[4];
#pragma unroll
    for (int s = 0; s < 4; ++s)
#pragma unroll
        for (int r = 0; r < 8; ++r) acc[s][r] = bsem[c0 + hi * 8 + r];

    __builtin_prefetch(arow, 0, 3);
    for (int k = 0; k < CS; k += 4) {
        const v2f a = *(const v2f*)(arow + k);           // W[c0+m][k+2hi+{0,1}]
        const size_t ko = (size_t)(k + 2 * hi) * HWS;
#pragma unroll
        for (int s = 0; s < 4; ++s) {
            v2f b;
            b.x = fbase[s][ko];                          // feat[b][k'][y][x]
            b.y = fbase[s][ko + HWS];
            acc[s] = wmma4(a, b, acc[s]);
        }
    }
    // D layout: vgpr r, lane -> row (out-ch) = c0 + 8*hi + r, col (pixel)
#pragma unroll
    for (int s = 0; s < 4; ++s) {
        const int myPix = quad * 64 + s * 16 + m;
        float* outp = semt + (size_t)myPix * CS + c0 + 8 * hi;
        float4 s0, s1;
        s0.x = fmaxf(acc[s][0], 0.f); s0.y = fmaxf(acc[s][1], 0.f);
        s0.z = fmaxf(acc[s][2], 0.f); s0.w = fmaxf(acc[s][3], 0.f);
        s1.x = fmaxf(acc[s][4], 0.f); s1.y = fmaxf(acc[s][5], 0.f);
        s1.z = fmaxf(acc[s][6], 0.f); s1.w = fmaxf(acc[s][7], 0.f);
        *(float4*)(outp)     = s0;
        *(float4*)(outp + 4) = s1;
    }
}

// ---------------------------------------------------------------------------
// K2: per-instance single-channel preds + exact top-k(392 of 784) by ranking
// ---------------------------------------------------------------------------
__global__ void preds_topk_kernel(const float* __restrict__ feats,
                                  const float* __restrict__ Winst,
                                  const float* __restrict__ binst,
                                  const float* __restrict__ Wdet,
                                  const float* __restrict__ bdet,
                                  const int* __restrict__ labels,
                                  float* __restrict__ instOut,
                                  float* __restrict__ detOut,
                                  int* __restrict__ pidx) {
    __shared__ float wi[CC], wd[CC], vals[HW];
    const int n = blockIdx.x, t = threadIdx.x;
    const int lab = labels[n];
    wi[t] = Winst[(size_t)lab * CC + t];
    wd[t] = Wdet[(size_t)lab * CC + t];
    __syncthreads();
    const float* f = feats + (size_t)n * CC * HW;
    const float bi = binst[lab], bd = bdet[lab];
    for (int pix = t; pix < HW; pix += 256) {
        float ai = 0.f, ad = 0.f;
        for (int c = 0; c < CC; ++c) {
            float v = f[(size_t)c * HW + pix];
            ai += wi[c] * v;
            ad += wd[c] * v;
        }
        ai += bi; ad += bd;
        instOut[(size_t)n * HW + pix] = ai;
        detOut[(size_t)n * HW + pix]  = ad;
        vals[pix] = ai;                 // detail_masks = instance_preds
    }
    __syncthreads();
    for (int pix = t; pix < HW; pix += 256) {
        const float v = vals[pix];
        int r = 0;
        for (int j = 0; j < HW; ++j) {
            float u = vals[j];
            r += (u > v) || (u == v && j < pix);   // top_k stable order
        }
        if (r < PP) pidx[(size_t)n * PP + r] = pix;
    }
}

// ---------------------------------------------------------------------------
// K3: per selected point: bilinear sample of sem_t (zero outside) -> fine,
//     gather coarse point features. One block per point, thread = channel.
// ---------------------------------------------------------------------------
__global__ void sample_build_kernel(const float* __restrict__ semt,
                                    const float* __restrict__ feats,
                                    const float* __restrict__ rois,
                                    const int* __restrict__ pidx,
                                    float* __restrict__ fine,
                                    float* __restrict__ coarse) {
    const int np = blockIdx.x;
    const int n  = np / PP;
    const int p  = np - n * PP;
    const int c  = threadIdx.x;
    const int pi = pidx[(size_t)n * PP + p];
    const float* roi = rois + (size_t)n * 5;
    const int bImg = (int)roi[0];
    const float xs = ((float)(pi % WW) + 0.5f) / (float)WW;
    const float ys = ((float)(pi / WW) + 0.5f) / (float)HH;
    const float fx = (roi[1] + xs * (roi[3] - roi[1])) * 0.25f - 0.5f;
    const float fy = (roi[2] + ys * (roi[4] - roi[2])) * 0.25f - 0.5f;
    const float x0f = floorf(fx), y0f = floorf(fy);
    const int x0 = (int)x0f, y0 = (int)y0f;
    const float wx1 = fx - x0f, wx0 = 1.f - wx1;
    const float wy1 = fy - y0f, wy0 = 1.f - wy1;

    float acc = 0.f;
#pragma unroll
    for (int dy = 0; dy < 2; ++dy) {
#pragma unroll
        for (int dx = 0; dx < 2; ++dx) {
            const int xi = x0 + dx, yi = y0 + dy;
            const float w = (dx ? wx1 : wx0) * (dy ? wy1 : wy0);
            if (xi >= 0 && xi < WS && yi >= 0 && yi < HS)
                acc += w * semt[(((size_t)bImg * HS + yi) * WS + xi) * CS + c];
        }
    }
    fine[(size_t)np * CC + c]   = acc;
    coarse[(size_t)np * CC + c] = feats[((size_t)n * CC + c) * HW + pi];
}

// ---------------------------------------------------------------------------
// K4: MLP layer  out[pt][oc] = act( W[oc,0:256].in1[pt] + W[oc,256:512].in2[pt] + b[oc] )
// one wave -> 16(oc) x 64(pt) tile: 4 accumulators share each A fetch.
// waves = (256/16)*(75264/64) = 18816 ; 8 waves/block -> 2352 blocks
// ---------------------------------------------------------------------------
__global__ void mlp_gemm_kernel(const float* __restrict__ in1,
                                const float* __restrict__ in2,
                                const float* __restrict__ Wf,
                                const float* __restrict__ bf,
                                float* __restrict__ outb,
                                int doRelu) {
    const int lane = threadIdx.x & 31;
    const int wave = threadIdx.x >> 5;
    const int tile = blockIdx.x * 8 + wave;
    const int mt   = tile & 15;
    const int quad = tile >> 4;          // 64-point quad [0,1176)
    const int m  = lane & 15;
    const int hi = lane >> 4;
    const int c0 = mt * 16;
    const int koff = 2 * hi;

    const float* arow = Wf + (size_t)(c0 + m) * 512 + koff;
    const float* r1[4];
    const float* r2[4];
#pragma unroll
    for (int s = 0; s < 4; ++s) {
        const int myPt = quad * 64 + s * 16 + m;
        r1[s] = in1 + (size_t)myPt * CC;
        r2[s] = in2 + (size_t)myPt * CC;
    }

    v8f acc[4];
#pragma unroll
    for (int s = 0; s < 4; ++s)
#pragma unroll
        for (int r = 0; r < 8; ++r) acc[s][r] = bf[c0 + hi * 8 + r];

    __builtin_prefetch(arow + 256, 0, 3);
    for (int k = 0; k < 256; k += 4) {               // first half: varying part
        const v2f a = *(const v2f*)(arow + k);
#pragma unroll
        for (int s = 0; s < 4; ++s) {
            const v2f b = *(const v2f*)(r1[s] + k + koff);
            acc[s] = wmma4(a, b, acc[s]);
        }
    }
    for (int k = 0; k < 256; k += 4) {               // second half: coarse part
        const v2f a = *(const v2f*)(arow + 256 + k);
#pragma unroll
        for (int s = 0; s < 4; ++s) {
            const v2f b = *(const v2f*)(r2[s] + k + koff);
            acc[s] = wmma4(a, b, acc[s]);
        }
    }
#pragma unroll
    for (int s = 0; s < 4; ++s) {
        const int myPt = quad * 64 + s * 16 + m;
        float* outp = outb + (size_t)myPt * CC + c0 + 8 * hi;
        float4 s0, s1;
        if (doRelu) {
            s0.x = fmaxf(acc[s][0], 0.f); s0.y = fmaxf(acc[s][1], 0.f);
            s0.z = fmaxf(acc[s][2], 0.f); s0.w = fmaxf(acc[s][3], 0.f);
            s1.x = fmaxf(acc[s][4], 0.f); s1.y = fmaxf(acc[s][5], 0.f);
            s1.z = fmaxf(acc[s][6], 0.f); s1.w = fmaxf(acc[s][7], 0.f);
        } else {
            s0.x = acc[s][0]; s0.y = acc[s][1]; s0.z = acc[s][2]; s0.w = acc[s][3];
            s1.x = acc[s][4]; s1.y = acc[s][5]; s1.z = acc[s][6]; s1.w = acc[s][7];
        }
        *(float4*)(outp)     = s0;
        *(float4*)(outp + 4) = s1;
    }
}

// ---------------------------------------------------------------------------
// K5: refined = instance_feats (float4 grid-stride copy)
// ---------------------------------------------------------------------------
__global__ void copy_kernel(const float4* __restrict__ src,
                            float4* __restrict__ dst, int n4) {
    for (int i = blockIdx.x * blockDim.x + threadIdx.x; i < n4;
         i += gridDim.x * blockDim.x)
        dst[i] = src[i];
}

// ---------------------------------------------------------------------------
// K6: scatter fuse output back into refined at selected points
// ---------------------------------------------------------------------------
__global__ void scatter_kernel(const float* __restrict__ fuseOut,
                               const int* __restrict__ pidx,
                               float* __restrict__ refined) {
    const int np = blockIdx.x;
    const int n  = np / PP;
    const int p  = np - n * PP;
    const int c  = threadIdx.x;
    const int pi = pidx[(size_t)n * PP + p];
    refined[((size_t)n * CC + c) * HW + pi] = fuseOut[(size_t)np * CC + c];
}

// ---------------------------------------------------------------------------
// K7: relu(refined) then 2x bilinear upsample (half-pixel centers, edge clamp)
// ---------------------------------------------------------------------------
__global__ void upsample_kernel(const float* __restrict__ refined,
                                float* __restrict__ out) {
    const long long i = (long long)blockIdx.x * blockDim.x + threadIdx.x;
    const int OW = 2 * WW, OH = 2 * HH;
    const int ow = (int)(i % OW);
    long long t = i / OW;
    const int oh = (int)(t % OH); t /= OH;
    const int cc = (int)(t % CC);
    const int n  = (int)(t / CC);

    const int hm = oh >> 1, wm = ow >> 1;
    const int hodd = oh & 1, wodd = ow & 1;
    const int h0 = hodd ? hm : hm - 1;
    const int w0 = wodd ? wm : wm - 1;
    const float fh = hodd ? 0.25f : 0.75f;    // weight of h0+1
    const float fw = wodd ? 0.25f : 0.75f;    // weight of w0+1
    const int h0c = h0 < 0 ? 0 : h0;
    const int h1c = (h0 + 1) > (HH - 1) ? (HH - 1) : (h0 + 1);
    const int w0c = w0 < 0 ? 0 : w0;
    const int w1c = (w0 + 1) > (WW - 1) ? (WW - 1) : (w0 + 1);

    const float* base = refined + ((size_t)n * CC + cc) * HW;
    const float v00 = fmaxf(base[h0c * WW + w0c], 0.f);
    const float v01 = fmaxf(base[h0c * WW + w1c], 0.f);
    const float v10 = fmaxf(base[h1c * WW + w0c], 0.f);
    const float v11 = fmaxf(base[h1c * WW + w1c], 0.f);
    out[i] = (v00 * (1.f - fw) + v01 * fw) * (1.f - fh) +
             (v10 * (1.f - fw) + v11 * fw) * fh;
}

// ---------------------------------------------------------------------------
extern "C" void kernel_launch(void* const* d_in, const int* in_sizes, int n_in,
                              void* d_out, int out_size, void* d_ws, size_t ws_size,
                              hipStream_t stream) {
    const float* feats   = (const float*)d_in[0];   // (192,256,28,28)
    const float* semfeat = (const float*)d_in[1];   // (2,256,200,304)
    // d_in[2] semantic_pred: unused by reference
    const float* rois    = (const float*)d_in[3];   // (192,5)
    const int*   labels  = (const int*)d_in[4];     // (192,)
    const float* W_sem   = (const float*)d_in[5];
    const float* b_sem   = (const float*)d_in[6];
    const float* W_inst  = (const float*)d_in[7];
    const float* b_inst  = (const float*)d_in[8];
    const float* W_det   = (const float*)d_in[9];
    const float* b_det   = (const float*)d_in[10];
    const float* W_fc1   = (const float*)d_in[11];
    const float* b_fc1   = (const float*)d_in[12];
    const float* W_fc2   = (const float*)d_in[13];
    const float* b_fc2   = (const float*)d_in[14];
    const float* W_fc3   = (const float*)d_in[15];
    const float* b_fc3   = (const float*)d_in[16];
    const float* W_fuse  = (const float*)d_in[17];
    const float* b_fuse  = (const float*)d_in[18];

    float* instOut = (float*)d_out;                 // 192*784
    float* detOut  = instOut + (size_t)NN * HW;     // 192*784
    float* upOut   = detOut + (size_t)NN * HW;      // 192*256*56*56

    // workspace layout (floats); sem_t region reused as `refined` after K3
    float* semt    = (float*)d_ws;                                  // 38,535,168 (max of sem_t / refined)
    float* bufA    = semt + (size_t)NN * CC * HW;                   // 19,267,584
    float* bufB    = bufA + (size_t)NP * CC;                        // 19,267,584
    float* coarse  = bufB + (size_t)NP * CC;                        // 19,267,584
    int*   pidx    = (int*)(coarse + (size_t)NP * CC);              // 75,264 ints
    float* refined = semt;

    // K1: semantic 1x1 conv + relu (WMMA f32, 16x64 tiles per wave)
    sem_gemm_kernel<<<(16 * (PIX / 64)) / 8, 256, 0, stream>>>(semfeat, W_sem, b_sem, semt);

    // K2: per-label preds + top-k point selection
    preds_topk_kernel<<<NN, 256, 0, stream>>>(feats, W_inst, b_inst, W_det, b_det,
                                              labels, instOut, detOut, pidx);

    // K3: bilinear sample + coarse gather (consumes sem_t)
    sample_build_kernel<<<NP, 256, 0, stream>>>(semt, feats, rois, pidx, bufA, coarse);

    // K4: 4-layer point MLP (WMMA f32, 16x64 tiles), ping-pong, split-K concat
    const int mlpBlocks = (16 * (NP / 64)) / 8;  // 2352
    mlp_gemm_kernel<<<mlpBlocks, 256, 0, stream>>>(bufA, coarse, W_fc1, b_fc1, bufB, 1);
    mlp_gemm_kernel<<<mlpBlocks, 256, 0, stream>>>(bufB, coarse, W_fc2, b_fc2, bufA, 1);
    mlp_gemm_kernel<<<mlpBlocks, 256, 0, stream>>>(bufA, coarse, W_fc3, b_fc3, bufB, 1);
    mlp_gemm_kernel<<<mlpBlocks, 256, 0, stream>>>(bufB, coarse, W_fuse, b_fuse, bufA, 0);

    // K5: refined = instance_feats   (sem_t region now dead -> reuse)
    const int n4 = (NN * CC * HW) / 4;
    copy_kernel<<<4096, 256, 0, stream>>>((const float4*)feats, (float4*)refined, n4);

    // K6: scatter fuse output into refined
    scatter_kernel<<<NP, 256, 0, stream>>>(bufA, pidx, refined);

    // K7: relu + 2x bilinear upsample -> final output
    const long long total = (long long)NN * CC * (2 * HH) * (2 * WW);
    upsample_kernel<<<(unsigned)(total / 256), 256, 0, stream>>>(refined, upOut);
}